// GATBlock_11656541241822
// MI455X (gfx1250) — compile-verified
//
#include <hip/hip_runtime.h>

#define IN_F   128
#define OUT_F  256
#define HEADS  4
#define DHEAD  64
#define NEG_SLOPE 0.2f
#define BN_EPS 1e-5f
#define APAD   132   // padded LDS row stride (floats): rows 16B-aligned, ds conflict-free

typedef __attribute__((ext_vector_type(2))) float v2f;
typedef __attribute__((ext_vector_type(8))) float v8f;
typedef __attribute__((ext_vector_type(4))) int   i32x4;

#if __has_builtin(__builtin_amdgcn_global_load_async_to_lds_b128) && \
    __has_builtin(__builtin_amdgcn_s_wait_asynccnt)
#define USE_ASYNC_LDS 1
typedef __attribute__((address_space(1))) i32x4* gvec_p;   // global int4*
typedef __attribute__((address_space(3))) i32x4* lvec_p;   // LDS int4*
#else
#define USE_ASYNC_LDS 0
#endif

// ---------- helpers ----------
__device__ __forceinline__ void atomicMaxF(float* addr, float v) {
    // valid for any sign mix when *addr initialized to -inf
    if (v >= 0.0f) atomicMax((int*)addr, __float_as_int(v));
    else           atomicMin((unsigned int*)addr, __float_as_uint(v));
}

__device__ __forceinline__ float leaky(float e) {
    return e > 0.0f ? e : e * NEG_SLOPE;
}

// Stage a 16-row x 128-col f32 A tile into LDS (8 KB), 512 x 16B chunks.
__device__ __forceinline__ void load_stripe(const float* __restrict__ A, int row0, int M,
                                            float (*As)[APAD]) {
    int t = threadIdx.x;
    #pragma unroll
    for (int i = 0; i < 2; ++i) {
        int c   = t + i * 256;          // chunk id 0..511
        int row = c >> 5;               // 32 chunks per row
        int kc  = c & 31;
        int grow = row0 + row; if (grow >= M) grow = M - 1;
        const float* gsrc = A + (size_t)grow * IN_F + kc * 4;
        float* ldst = &As[row][kc * 4]; // byte addr row*528 + kc*16 -> 16B aligned
#if USE_ASYNC_LDS
        __builtin_amdgcn_global_load_async_to_lds_b128(
            (gvec_p)gsrc, (lvec_p)ldst, 0, 0);
#else
        *(float4*)ldst = *(const float4*)gsrc;
#endif
    }
#if USE_ASYNC_LDS
    __builtin_amdgcn_s_wait_asynccnt(0);
#endif
    __syncthreads();
}

// Each wave computes two 16x16 tiles (cols nt0*16.. and +16) of A_stripe * B.
__device__ __forceinline__ void wave_tiles(const float (*As)[APAD], const float* __restrict__ B,
                                           int nt0, v8f& acc0, v8f& acc1) {
    int lane = threadIdx.x & 31;
    int r    = lane & 15;
    int kk   = (lane >> 4) << 1;        // lanes 0-15 -> K 0,1 ; 16-31 -> K 2,3
    int bcol0 = nt0 * 16 + r;
    int bcol1 = bcol0 + 16;
    #pragma unroll 4
    for (int k0 = 0; k0 < IN_F; k0 += 4) {
        v2f a, b0, b1;
        a.x = As[r][k0 + kk];           // ds_load_b64, conflict-free via APAD
        a.y = As[r][k0 + kk + 1];
        const float* Brow0 = B + (size_t)(k0 + kk)     * OUT_F;
        const float* Brow1 = B + (size_t)(k0 + kk + 1) * OUT_F;
        b0.x = Brow0[bcol0]; b0.y = Brow1[bcol0];
        b1.x = Brow0[bcol1]; b1.y = Brow1[bcol1];
        acc0 = __builtin_amdgcn_wmma_f32_16x16x4_f32(false, a, false, b0,
                                                     (short)0, acc0, false, false);
        acc1 = __builtin_amdgcn_wmma_f32_16x16x4_f32(false, a, false, b1,
                                                     (short)0, acc1, false, false);
    }
}

// ---------- init ----------
__global__ void k_init_out(float* __restrict__ out, const float* __restrict__ bias, int n) {
    int idx = blockIdx.x * blockDim.x + threadIdx.x;
    if (idx < n) out[idx] = bias[idx & (OUT_F - 1)];
}

__global__ void k_init_attn(float* __restrict__ m, float* __restrict__ s,
                            float* __restrict__ colsum, float* __restrict__ colsumsq, int nh) {
    int idx = blockIdx.x * blockDim.x + threadIdx.x;
    if (idx < nh) {
        m[idx] = __int_as_float(0xFF800000);  // -inf
        s[idx] = 0.0f;
    }
    if (idx < OUT_F) {
        colsum[idx] = 0.0f;
        colsumsq[idx] = 0.0f;
    }
}

// ---------- h = x @ W : block = one 16-row stripe, LDS-staged A, WMMA fp32 ----------
__global__ __launch_bounds__(256)
void k_gemm(const float* __restrict__ A, const float* __restrict__ B,
            float* __restrict__ C, int M) {
    __shared__ float As[16][APAD];
    int row0 = blockIdx.x * 16;
    load_stripe(A, row0, M, As);

    int nt0 = (threadIdx.x >> 5) * 2;   // wave id * 2 column tiles
    v8f acc0 = {}, acc1 = {};
    wave_tiles(As, B, nt0, acc0, acc1);

    int lane  = threadIdx.x & 31;
    int mbase = row0 + ((lane >> 4) << 3);
    int col0  = nt0 * 16 + (lane & 15);
    #pragma unroll
    for (int i = 0; i < 8; ++i) {
        int row = mbase + i;
        if (row < M) {
            C[(size_t)row * OUT_F + col0]      = acc0[i];
            C[(size_t)row * OUT_F + col0 + 16] = acc1[i];
        }
    }
}

// ---------- per-(node,head) attention scalars ----------
__global__ void k_attn_vec(const float* __restrict__ h,
                           const float* __restrict__ att_src,
                           const float* __restrict__ att_dst,
                           float* __restrict__ a_src, float* __restrict__ a_dst, int nh) {
    int idx = blockIdx.x * blockDim.x + threadIdx.x;
    if (idx >= nh) return;
    int n  = idx >> 2;
    int hh = idx & 3;
    const float* hv = h + (size_t)n * OUT_F + hh * DHEAD;
    const float* ws = att_src + hh * DHEAD;
    const float* wd = att_dst + hh * DHEAD;
    float as = 0.0f, ad = 0.0f;
    #pragma unroll 8
    for (int d = 0; d < DHEAD; ++d) {
        float v = hv[d];
        as += v * ws[d];
        ad += v * wd[d];
    }
    a_src[idx] = as;
    a_dst[idx] = ad;
}

// ---------- softmax pass 1: per-target max ----------
__global__ void k_edge_max(const long long* __restrict__ ei,
                           const float* __restrict__ a_src, const float* __restrict__ a_dst,
                           float* __restrict__ m, int E, int N) {
    long long idx = (long long)blockIdx.x * blockDim.x + threadIdx.x;
    long long total = (long long)(E + N) * HEADS;
    if (idx >= total) return;
    int edge = (int)(idx >> 2);
    int hh   = (int)(idx & 3);
    int src, dst;
    if (edge < E) { src = (int)ei[edge]; dst = (int)ei[E + edge]; }
    else          { src = dst = edge - E; }
    float e = leaky(a_src[src * HEADS + hh] + a_dst[dst * HEADS + hh]);
    atomicMaxF(&m[dst * HEADS + hh], e);
}

__global__ void k_clamp_m(float* __restrict__ m, int nh) {
    int idx = blockIdx.x * blockDim.x + threadIdx.x;
    if (idx >= nh) return;
    float v = m[idx];
    if (!(v > __int_as_float(0xFF800000))) m[idx] = 0.0f;  // still -inf -> 0
}

// ---------- softmax pass 2: per-target exp-sum ----------
__global__ void k_edge_sum(const long long* __restrict__ ei,
                           const float* __restrict__ a_src, const float* __restrict__ a_dst,
                           const float* __restrict__ m, float* __restrict__ s, int E, int N) {
    long long idx = (long long)blockIdx.x * blockDim.x + threadIdx.x;
    long long total = (long long)(E + N) * HEADS;
    if (idx >= total) return;
    int edge = (int)(idx >> 2);
    int hh   = (int)(idx & 3);
    int src, dst;
    if (edge < E) { src = (int)ei[edge]; dst = (int)ei[E + edge]; }
    else          { src = dst = edge - E; }
    float e  = leaky(a_src[src * HEADS + hh] + a_dst[dst * HEADS + hh]);
    float ex = __expf(e - m[dst * HEADS + hh]);
    atomicAdd(&s[dst * HEADS + hh], ex);
}

// ---------- softmax pass 3: weighted scatter (one wave32 per edge) ----------
__global__ __launch_bounds__(256)
void k_scatter(const long long* __restrict__ ei, const float* __restrict__ h,
               const float* __restrict__ a_src, const float* __restrict__ a_dst,
               const float* __restrict__ m, const float* __restrict__ s,
               float* __restrict__ out, int E, int N) {
    int edge = blockIdx.x * (blockDim.x >> 5) + (threadIdx.x >> 5);
    if (edge >= E + N) return;
    int lane = threadIdx.x & 31;
    int src, dst;
    if (edge < E) { src = (int)ei[edge]; dst = (int)ei[E + edge]; }
    else          { src = dst = edge - E; }
    float alpha[HEADS];
    #pragma unroll
    for (int hh = 0; hh < HEADS; ++hh) {
        float e  = leaky(a_src[src * HEADS + hh] + a_dst[dst * HEADS + hh]);
        float ex = __expf(e - m[dst * HEADS + hh]);
        alpha[hh] = ex / (s[dst * HEADS + hh] + 1e-16f);
    }
    const float* hs = h + (size_t)src * OUT_F;   // L2-resident (102 MB < 192 MB L2)
    float* od = out + (size_t)dst * OUT_F;
    #pragma unroll
    for (int j = 0; j < OUT_F / 32; ++j) {
        int f = lane + 32 * j;
        atomicAdd(&od[f], hs[f] * alpha[f >> 6]);
    }
}

// ---------- BN column stats ----------
__global__ void k_colstats(const float* __restrict__ out, float* __restrict__ colsum,
                           float* __restrict__ colsumsq, int N, int rowsPer) {
    int col = threadIdx.x;                 // 256 threads = 256 columns
    int r0 = blockIdx.x * rowsPer;
    int r1 = r0 + rowsPer; if (r1 > N) r1 = N;
    float su = 0.0f, sq = 0.0f;
    for (int r = r0; r < r1; ++r) {
        float v = out[(size_t)r * OUT_F + col];
        su += v; sq += v * v;
    }
    atomicAdd(&colsum[col], su);
    atomicAdd(&colsumsq[col], sq);
}

__global__ void k_statsfin(float* __restrict__ colsum, float* __restrict__ colsumsq, int N) {
    int c = threadIdx.x;
    float inv = 1.0f / (float)N;
    float mu  = colsum[c] * inv;
    float var = colsumsq[c] * inv - mu * mu;
    colsum[c]   = mu;                        // reuse as mean
    colsumsq[c] = rsqrtf(var + BN_EPS);      // reuse as inv_std
}

// ---------- fused: residual proj (WMMA, LDS-staged x) + BN + ReLU, in place ----------
__global__ __launch_bounds__(256)
void k_final(const float* __restrict__ x, const float* __restrict__ pW,
             const float* __restrict__ pb, const float* __restrict__ gamma,
             const float* __restrict__ beta, const float* __restrict__ mean,
             const float* __restrict__ istd, float* __restrict__ out, int M) {
    __shared__ float As[16][APAD];
    int row0 = blockIdx.x * 16;
    load_stripe(x, row0, M, As);

    int nt0 = (threadIdx.x >> 5) * 2;
    v8f acc0 = {}, acc1 = {};
    wave_tiles(As, pW, nt0, acc0, acc1);

    int lane  = threadIdx.x & 31;
    int mbase = row0 + ((lane >> 4) << 3);
    int col0  = nt0 * 16 + (lane & 15);
    #pragma unroll
    for (int t = 0; t < 2; ++t) {
        int col = col0 + t * 16;
        float mu = mean[col], is = istd[col], ga = gamma[col];
        float be = beta[col], pbi = pb[col];
        #pragma unroll
        for (int i = 0; i < 8; ++i) {
            int row = mbase + i;
            if (row >= M) continue;
            size_t o = (size_t)row * OUT_F + col;
            float g = out[o];                           // gat + bias (accumulated)
            float a = (t == 0) ? acc0[i] : acc1[i];
            float v = (g - mu) * is * ga + be + a + pbi;
            out[o] = v > 0.0f ? v : 0.0f;
        }
    }
}

extern "C" void kernel_launch(void* const* d_in, const int* in_sizes, int n_in,
                              void* d_out, int out_size, void* d_ws, size_t ws_size,
                              hipStream_t stream) {
    const float*     x       = (const float*)d_in[0];
    const long long* ei      = (const long long*)d_in[1];   // int64 [2,E]
    const float*     W       = (const float*)d_in[2];
    const float*     att_src = (const float*)d_in[3];
    const float*     att_dst = (const float*)d_in[4];
    const float*     bias    = (const float*)d_in[5];
    const float*     gamma   = (const float*)d_in[6];
    const float*     beta    = (const float*)d_in[7];
    const float*     proj_W  = (const float*)d_in[8];
    const float*     proj_b  = (const float*)d_in[9];
    float*           out     = (float*)d_out;

    const int N = in_sizes[0] / IN_F;
    const int E = in_sizes[1] / 2;
    const int EN = E + N;

    // workspace carving (floats)
    float* ws = (float*)d_ws;
    size_t off = 0;
    float* hbuf     = ws + off; off += (size_t)N * OUT_F;   // 102.4 MB
    float* a_src    = ws + off; off += (size_t)N * HEADS;
    float* a_dst    = ws + off; off += (size_t)N * HEADS;
    float* mbuf     = ws + off; off += (size_t)N * HEADS;
    float* sbuf     = ws + off; off += (size_t)N * HEADS;
    float* colsum   = ws + off; off += OUT_F;
    float* colsumsq = ws + off; off += OUT_F;
    (void)ws_size; (void)n_in; (void)out_size;

    const int T = 256;
    const int NH = N * HEADS;

    // 1. d_out <- bias (gat accumulator), m <- -inf, s <- 0, stats <- 0
    k_init_out<<<((size_t)N * OUT_F + T - 1) / T, T, 0, stream>>>(out, bias, N * OUT_F);
    k_init_attn<<<(NH + T - 1) / T, T, 0, stream>>>(mbuf, sbuf, colsum, colsumsq, NH);

    // 2. h = x @ W   (WMMA fp32, one block per 16-row stripe)
    int stripes = (N + 15) / 16;
    k_gemm<<<stripes, T, 0, stream>>>(x, W, hbuf, N);

    // 3. per-(node,head) attention scalars
    k_attn_vec<<<(NH + T - 1) / T, T, 0, stream>>>(hbuf, att_src, att_dst, a_src, a_dst, NH);

    // 4-6. softmax over incoming edges (max, clamp, exp-sum)
    long long enh = (long long)EN * HEADS;
    k_edge_max<<<(unsigned)((enh + T - 1) / T), T, 0, stream>>>(ei, a_src, a_dst, mbuf, E, N);
    k_clamp_m<<<(NH + T - 1) / T, T, 0, stream>>>(mbuf, NH);
    k_edge_sum<<<(unsigned)((enh + T - 1) / T), T, 0, stream>>>(ei, a_src, a_dst, mbuf, sbuf, E, N);

    // 7. weighted scatter-add (one wave32 per edge)
    k_scatter<<<(EN + 7) / 8, T, 0, stream>>>(ei, hbuf, a_src, a_dst, mbuf, sbuf, out, E, N);

    // 8-9. BatchNorm batch statistics
    int nblk = 512;
    int rowsPer = (N + nblk - 1) / nblk;
    k_colstats<<<nblk, T, 0, stream>>>(out, colsum, colsumsq, N, rowsPer);
    k_statsfin<<<1, T, 0, stream>>>(colsum, colsumsq, N);

    // 10. fused BN + residual projection (WMMA) + ReLU, in place
    k_final<<<stripes, T, 0, stream>>>(x, proj_W, proj_b, gamma, beta,
                                       colsum, colsumsq, out, N);
}